// LiftSplat_75728863363275
// MI455X (gfx1250) — compile-verified
//
#include <hip/hip_runtime.h>

// ---- static config (mirrors reference) ----
#define Bc   4
#define Nc   4
#define Cc   64
#define Dc   41
#define FHc  16
#define FWc  44
#define NXc  240
#define NYc  240
#define NZc  1
#define PTS_PER_BN (Dc * FHc * FWc)        // 28864, divisible by 32
#define NPTS       (Bc * Nc * PTS_PER_BN)  // 461824
#define YXc        (NYc * NXc)             // 57600

typedef float v2f __attribute__((ext_vector_type(2)));
typedef float v8f __attribute__((ext_vector_type(8)));

#if defined(__HIP_DEVICE_COMPILE__) && defined(__gfx1250__) && \
    __has_builtin(__builtin_amdgcn_wmma_f32_16x16x4_f32)
#define USE_WMMA 1
#else
#define USE_WMMA 0
#endif

// ---------------------------------------------------------------------------
// 3x3 inverse via adjugate
// ---------------------------------------------------------------------------
__device__ __forceinline__ void inv3x3(const float* m, float* o) {
    float a = m[0], b = m[1], c = m[2];
    float d = m[3], e = m[4], f = m[5];
    float g = m[6], h = m[7], i = m[8];
    float A =  (e * i - f * h);
    float B = -(d * i - f * g);
    float C =  (d * h - e * g);
    float det = a * A + b * B + c * C;
    float r = 1.0f / det;
    o[0] = A * r;
    o[1] = -(b * i - c * h) * r;
    o[2] =  (b * f - c * e) * r;
    o[3] = B * r;
    o[4] =  (a * i - c * g) * r;
    o[5] = -(a * f - c * d) * r;
    o[6] = C * r;
    o[7] = -(a * h - b * g) * r;
    o[8] =  (a * e - b * d) * r;
}

// Per-(b,n) packed constants: [invPostRot 0..8][combine 9..17][trans 18..20][post_trans 21..23]
__global__ void prep_mats(const float* __restrict__ rots,
                          const float* __restrict__ trans,
                          const float* __restrict__ intrins,
                          const float* __restrict__ post_rots,
                          const float* __restrict__ post_trans,
                          float* __restrict__ mats) {
    int bn = threadIdx.x;
    if (bn >= Bc * Nc) return;
    float iPR[9], iK[9], CB[9];
    inv3x3(post_rots + bn * 9, iPR);
    inv3x3(intrins + bn * 9, iK);
    const float* R = rots + bn * 9;
    for (int i = 0; i < 3; ++i)
        for (int j = 0; j < 3; ++j)
            CB[i * 3 + j] = R[i * 3 + 0] * iK[0 * 3 + j] +
                            R[i * 3 + 1] * iK[1 * 3 + j] +
                            R[i * 3 + 2] * iK[2 * 3 + j];
    float* o = mats + bn * 24;
    for (int i = 0; i < 9; ++i) o[i]     = iPR[i];
    for (int i = 0; i < 9; ++i) o[9 + i] = CB[i];
    for (int i = 0; i < 3; ++i) o[18 + i] = trans[bn * 3 + i];
    for (int i = 0; i < 3; ++i) o[21 + i] = post_trans[bn * 3 + i];
}

// Truncating cast matches jnp .astype(int32) (trunc toward zero), incl. (-1,0) -> 0
__device__ __forceinline__ int lin_index(float px, float py, float pz, int b) {
    const float offx = -47.8f - 0.4f * 0.5f;   // bx - dx/2 folded in f32
    const float offy = -47.8f - 0.4f * 0.5f;
    const float offz =   0.0f - 20.0f * 0.5f;
    int vx = (int)((px - offx) / 0.4f);
    int vy = (int)((py - offy) / 0.4f);
    int vz = (int)((pz - offz) / 20.0f);
    bool kept = (vx >= 0) & (vx < NXc) & (vy >= 0) & (vy < NYc) &
                (vz >= 0) & (vz < NZc);
    return kept ? (b * YXc + vy * NXc + vx) : -1;
}

// ---------------------------------------------------------------------------
// Geometry: one wave = 32 points, final affine transform via V_WMMA_F32_16X16X4_F32
//   A (16x4): rows 0..2 = [combine | trans], rest 0
//   B (4x16): columns = 16 points (qx, qy, qz, 1)
//   D rows 0..2 (lanes 0-15, VGPRs 0..2) = transformed xyz
// ---------------------------------------------------------------------------
__global__ void geom_kernel(const float* __restrict__ mats, int* __restrict__ lin) {
    int tid  = blockIdx.x * blockDim.x + threadIdx.x;
    int lane = threadIdx.x & 31;
    int base = tid & ~31;                 // first of this wave's 32 points
    int bn   = base / PTS_PER_BN;         // uniform per wave
    const float* M = mats + bn * 24;
    float iPR0 = M[0], iPR1 = M[1], iPR2 = M[2];
    float iPR3 = M[3], iPR4 = M[4], iPR5 = M[5];
    float iPR6 = M[6], iPR7 = M[7], iPR8 = M[8];
    float CB0 = M[9],  CB1 = M[10], CB2 = M[11];
    float CB3 = M[12], CB4 = M[13], CB5 = M[14];
    float CB6 = M[15], CB7 = M[16], CB8 = M[17];
    float TR0 = M[18], TR1 = M[19], TR2 = M[20];
    float PT0 = M[21], PT1 = M[22], PT2 = M[23];
    int bidx = bn / Nc;

#if USE_WMMA
    v2f Bop0, Bop1;
#pragma unroll
    for (int step = 0; step < 2; ++step) {
        int q  = base + step * 16 + (lane & 15);
        int r  = q - bn * PTS_PER_BN;
        int wi = r % FWc;
        int t2 = r / FWc;
        int hi = t2 % FHc;
        int di = t2 / FHc;
        float u  = (float)((double)wi * (703.0 / 43.0));
        float vv = (float)hi * 17.0f;
        float dd = 4.0f + (float)di;
        float x0 = u - PT0, y0 = vv - PT1, z0 = dd - PT2;
        float qx = iPR0 * x0 + iPR1 * y0 + iPR2 * z0;
        float qy = iPR3 * x0 + iPR4 * y0 + iPR5 * z0;
        float qz = iPR6 * x0 + iPR7 * y0 + iPR8 * z0;
        qx *= qz;
        qy *= qz;
        v2f bv;
        bv.x = (lane < 16) ? qx : qz;
        bv.y = (lane < 16) ? qy : 1.0f;
        if (step == 0) Bop0 = bv; else Bop1 = bv;
    }
    // A operand: lane holds A[m][k] with m = lane%16, k = (lane<16 ? {0,1} : {2,3})
    int m = lane & 15;
    float c0 = (lane < 16) ? CB0 : CB2;
    float c1 = (lane < 16) ? CB3 : CB5;
    float c2 = (lane < 16) ? CB6 : CB8;
    float e0 = (lane < 16) ? CB1 : TR0;
    float e1 = (lane < 16) ? CB4 : TR1;
    float e2 = (lane < 16) ? CB7 : TR2;
    float a0 = (m == 0) ? c0 : (m == 1) ? c1 : (m == 2) ? c2 : 0.0f;
    float a1 = (m == 0) ? e0 : (m == 1) ? e1 : (m == 2) ? e2 : 0.0f;
    v2f A;
    A.x = a0;
    A.y = a1;
    v8f cz = {0.f, 0.f, 0.f, 0.f, 0.f, 0.f, 0.f, 0.f};
    v8f d0 = __builtin_amdgcn_wmma_f32_16x16x4_f32(false, A, false, Bop0,
                                                   (short)0, cz, false, false);
    v8f d1 = __builtin_amdgcn_wmma_f32_16x16x4_f32(false, A, false, Bop1,
                                                   (short)0, cz, false, false);
    if (lane < 16) {
        lin[base + lane]      = lin_index(d0[0], d0[1], d0[2], bidx);
        lin[base + 16 + lane] = lin_index(d1[0], d1[1], d1[2], bidx);
    }
#else
    // Scalar fallback: one point per thread
    int q  = tid;
    int r  = q - bn * PTS_PER_BN;
    int wi = r % FWc;
    int t2 = r / FWc;
    int hi = t2 % FHc;
    int di = t2 / FHc;
    float u  = (float)((double)wi * (703.0 / 43.0));
    float vv = (float)hi * 17.0f;
    float dd = 4.0f + (float)di;
    float x0 = u - PT0, y0 = vv - PT1, z0 = dd - PT2;
    float qx = iPR0 * x0 + iPR1 * y0 + iPR2 * z0;
    float qy = iPR3 * x0 + iPR4 * y0 + iPR5 * z0;
    float qz = iPR6 * x0 + iPR7 * y0 + iPR8 * z0;
    qx *= qz;
    qy *= qz;
    float fx = CB0 * qx + CB1 * qy + CB2 * qz + TR0;
    float fy = CB3 * qx + CB4 * qy + CB5 * qz + TR1;
    float fz = CB6 * qx + CB7 * qy + CB8 * qz + TR2;
    lin[q] = lin_index(fx, fy, fz, bidx);
#endif
}

// ---------------------------------------------------------------------------
// Scatter: thread = (point, 4 channels). b128 coalesced load, 4 f32 L2 atomics
// into final [B, C, Y, X] layout (output fits in 192MB L2).
// ---------------------------------------------------------------------------
__global__ void scatter_kernel(const float4* __restrict__ x4,
                               const int* __restrict__ lin,
                               float* __restrict__ out) {
    int t  = blockIdx.x * blockDim.x + threadIdx.x;   // < NPTS*16
    int p  = t >> 4;
    int cq = (t & 15) << 2;                            // channel base
    float4 f = x4[t];
    int L = lin[p];
    if (L < 0) return;
    int bb = L / YXc;
    int sp = L - bb * YXc;
    float* o = out + (size_t)(bb * Cc + cq) * YXc + sp;
    atomicAdd(o,            f.x);
    atomicAdd(o + YXc,      f.y);
    atomicAdd(o + 2 * YXc,  f.z);
    atomicAdd(o + 3 * YXc,  f.w);
}

extern "C" void kernel_launch(void* const* d_in, const int* in_sizes, int n_in,
                              void* d_out, int out_size, void* d_ws, size_t ws_size,
                              hipStream_t stream) {
    const float* x          = (const float*)d_in[0];
    const float* rots       = (const float*)d_in[1];
    const float* trans      = (const float*)d_in[2];
    const float* intrins    = (const float*)d_in[3];
    const float* post_rots  = (const float*)d_in[4];
    const float* post_trans = (const float*)d_in[5];
    float* out = (float*)d_out;

    float* mats = (float*)d_ws;                         // 16*24 floats
    int*   lin  = (int*)((char*)d_ws + 2048);           // NPTS int32 (~1.8 MB)

    hipMemsetAsync(d_out, 0, (size_t)out_size * sizeof(float), stream);
    prep_mats<<<1, 32, 0, stream>>>(rots, trans, intrins, post_rots, post_trans, mats);
    geom_kernel<<<NPTS / 256, 256, 0, stream>>>(mats, lin);
    scatter_kernel<<<(NPTS * 16) / 256, 256, 0, stream>>>((const float4*)x, lin, out);
}